// SoftDiceLoss_5007931867553
// MI455X (gfx1250) — compile-verified
//
#include <hip/hip_runtime.h>
#include <hip/hip_bf16.h>
#include <stdint.h>

#define NUM_CLASSES 20
#define EPSF 1e-10f

#define THREADS 256
#define BLOCKS 2048
#define WAVES_PER_BLOCK (THREADS / 32)
#define HSTRIDE 33                 // padded sub-hist stride (odd -> bank decorrelation)
#define HSET (3 * HSTRIDE)         // 99 words per replica: [pred | tgt | tp]
#define REPLICAS 32                // one replica per lane -> no intra-wave same-address conflicts
#define STAGE_BYTES 1024           // per wave per buffer per tensor: 64 chunks * 16 B
#define CHUNKS_PER_WAVE_STAGE 64   // two 32-chunk tiles (second via offset:512)

// ---- CDNA5 async global->LDS copy (GLOBAL_LOAD_ASYNC_TO_LDS_B128, ASYNCcnt). ----
// Note: the IOFFSET immediate is added to BOTH the global and the LDS address
// (CDNA5 ISA), so offset:512 reuses the same address VGPRs for
// the second 512B tile of a stage with zero extra address math.
__device__ __forceinline__ void async_copy16(uint32_t lds_byte_off, uint64_t gaddr) {
    asm volatile("global_load_async_to_lds_b128 %0, %1, off"
                 :: "v"(lds_byte_off), "v"(gaddr) : "memory");
}
__device__ __forceinline__ void async_copy16_off512(uint32_t lds_byte_off, uint64_t gaddr) {
    asm volatile("global_load_async_to_lds_b128 %0, %1, off offset:512"
                 :: "v"(lds_byte_off), "v"(gaddr) : "memory");
}
__device__ __forceinline__ void wait_async_le4() {
    asm volatile("s_wait_asynccnt 0x4" ::: "memory");
}
__device__ __forceinline__ void wait_async_0() {
    asm volatile("s_wait_asynccnt 0x0" ::: "memory");
}

__device__ __forceinline__ void hist_accum(unsigned* hist, uint32_t hbase, int p, int t) {
    p &= 31; t &= 31;  // data contract is 0..19; clamp protects LDS layout from garbage
    atomicAdd(&hist[hbase + p], 1u);                     // pred count
    atomicAdd(&hist[hbase + HSTRIDE + t], 1u);           // target count
    if (p == t) atomicAdd(&hist[hbase + 2 * HSTRIDE + p], 1u);  // tp (EXEC-predicated, ~5%)
}

__device__ __forceinline__ void hist_accum4(unsigned* hist, uint32_t hbase, int4 p4, int4 t4) {
    hist_accum(hist, hbase, p4.x, t4.x);
    hist_accum(hist, hbase, p4.y, t4.y);
    hist_accum(hist, hbase, p4.z, t4.z);
    hist_accum(hist, hbase, p4.w, t4.w);
}

__global__ void __launch_bounds__(THREADS)
dice_hist_kernel(const int* __restrict__ pred, const int* __restrict__ tgt,
                 unsigned* __restrict__ ghist, long long nchunks, int tail) {
    __shared__ unsigned hist[REPLICAS * HSET];                                   // 12672 B
    __shared__ __align__(16) uint8_t stage[WAVES_PER_BLOCK * 2 * 2 * STAGE_BYTES]; // 32768 B

    for (int i = threadIdx.x; i < REPLICAS * HSET; i += blockDim.x) hist[i] = 0u;
    __syncthreads();

    const int lane = threadIdx.x & 31;
    const int wave = threadIdx.x >> 5;
    const uint32_t hbase = (uint32_t)lane * HSET;  // replica = lane

    const unsigned wavesTotal = gridDim.x * WAVES_PER_BLOCK;
    const unsigned waveId     = blockIdx.x * WAVES_PER_BLOCK + wave;
    const long long stageSpan = (long long)wavesTotal * CHUNKS_PER_WAVE_STAGE; // chunks/stage, device-wide
    const long long main_chunks = nchunks - (nchunks % stageSpan);
    const unsigned nstages = (unsigned)(main_chunks / stageSpan);              // wave-uniform

    // LDS byte offsets of this lane's staging slots: [wave][buf][tensor]
    const uint32_t stage_lds = (uint32_t)(uintptr_t)&stage[0];
    uint32_t slotP[2], slotT[2];
#pragma unroll
    for (int b = 0; b < 2; ++b) {
        slotP[b] = stage_lds + (uint32_t)(((wave * 2 + b) * 2 + 0) * STAGE_BYTES + lane * 16);
        slotT[b] = stage_lds + (uint32_t)(((wave * 2 + b) * 2 + 1) * STAGE_BYTES + lane * 16);
    }

    if (nstages > 0) {
        // Lane's running global addresses (chunk c0 = waveId*64 + lane; +32 chunks via offset:512)
        const uint64_t step = (uint64_t)stageSpan * 16u;
        uint64_t addrP = (uint64_t)(uintptr_t)pred +
                         ((uint64_t)waveId * CHUNKS_PER_WAVE_STAGE + (unsigned)lane) * 16u;
        uint64_t addrT = (uint64_t)(uintptr_t)tgt +
                         ((uint64_t)waveId * CHUNKS_PER_WAVE_STAGE + (unsigned)lane) * 16u;

        int buf = 0;
        // Prologue: issue stage 0 (4 async ops)
        async_copy16(slotP[0], addrP); async_copy16_off512(slotP[0], addrP);
        async_copy16(slotT[0], addrT); async_copy16_off512(slotT[0], addrT);
        addrP += step; addrT += step;

        for (unsigned k = 0; k < nstages; ++k) {
            const bool has_next = (k + 1u < nstages);  // wave-uniform
            if (has_next) {
                const int nb = buf ^ 1;
                async_copy16(slotP[nb], addrP); async_copy16_off512(slotP[nb], addrP);
                async_copy16(slotT[nb], addrT); async_copy16_off512(slotT[nb], addrT);
                addrP += step; addrT += step;
                wait_async_le4();   // newest 4 may be in flight; current buffer complete
            } else {
                wait_async_0();
            }
            const uint8_t* sp = &stage[((wave * 2 + buf) * 2 + 0) * STAGE_BYTES + lane * 16];
            const uint8_t* st = &stage[((wave * 2 + buf) * 2 + 1) * STAGE_BYTES + lane * 16];
            int4 p4a = *reinterpret_cast<const int4*>(sp);         // ds_load_b128
            int4 t4a = *reinterpret_cast<const int4*>(st);
            int4 p4b = *reinterpret_cast<const int4*>(sp + 512);
            int4 t4b = *reinterpret_cast<const int4*>(st + 512);
            hist_accum4(hist, hbase, p4a, t4a);
            hist_accum4(hist, hbase, p4b, t4b);
            buf ^= 1;
        }
    }

    // Ragged remainder (< one device stage of chunks): direct global_load_b128
    {
        const int4* __restrict__ pred4 = (const int4*)pred;
        const int4* __restrict__ tgt4  = (const int4*)tgt;
        const long long totalThreads = (long long)gridDim.x * blockDim.x;
        const long long gl = (long long)blockIdx.x * blockDim.x + threadIdx.x;
        for (long long r = main_chunks + gl; r < nchunks; r += totalThreads) {
            int4 p4 = pred4[r];
            int4 t4 = tgt4[r];
            hist_accum4(hist, hbase, p4, t4);
        }
    }
    // Scalar tail (N % 4), block 0 only
    if (blockIdx.x == 0 && threadIdx.x < tail) {
        const long long e = nchunks * 4 + threadIdx.x;
        hist_accum(hist, hbase, pred[e], tgt[e]);
    }

    __syncthreads();
    // Reduce 32 replicas -> global histogram (layout: [0..19]=pred, [32..51]=tgt, [64..83]=tp)
    for (int j = threadIdx.x; j < 3 * NUM_CLASSES; j += blockDim.x) {
        const int h = j / NUM_CLASSES, c = j % NUM_CLASSES;
        unsigned s = 0;
        for (int r = 0; r < REPLICAS; ++r) s += hist[r * HSET + h * HSTRIDE + c];
        atomicAdd(&ghist[h * 32 + c], s);
    }
}

__global__ void dice_zero_kernel(unsigned* g) {
    if (threadIdx.x < 96) g[threadIdx.x] = 0u;
}

__global__ void dice_finalize_kernel(const unsigned* __restrict__ g, float* __restrict__ out) {
    if (blockIdx.x == 0 && threadIdx.x == 0) {
        float s = 0.0f;
        for (int c = 1; c < NUM_CLASSES; ++c) {
            const float pc = (float)g[c];
            const float tc = (float)g[32 + c];
            const float tp = (float)g[64 + c];
            s += (2.0f * tp) / (pc + tc + EPSF);
        }
        out[0] = s / (float)(NUM_CLASSES - 1);
    }
}

extern "C" void kernel_launch(void* const* d_in, const int* in_sizes, int n_in,
                              void* d_out, int out_size, void* d_ws, size_t ws_size,
                              hipStream_t stream) {
    const int* pred = (const int*)d_in[0];
    const int* tgt  = (const int*)d_in[1];
    unsigned* gh = (unsigned*)d_ws;  // 96 words
    const long long N = (long long)in_sizes[0];
    const long long nchunks = N >> 2;
    const int tail = (int)(N & 3);

    dice_zero_kernel<<<1, 128, 0, stream>>>(gh);
    dice_hist_kernel<<<BLOCKS, THREADS, 0, stream>>>(pred, tgt, gh, nchunks, tail);
    dice_finalize_kernel<<<1, 64, 0, stream>>>(gh, (float*)d_out);
}